// GCNModel_45870250721504
// MI455X (gfx1250) — compile-verified
//
#include <hip/hip_runtime.h>
#include <stdint.h>

// ---- CDNA5 vector types ----
typedef float v2f __attribute__((ext_vector_type(2)));
typedef float v8f __attribute__((ext_vector_type(8)));
typedef unsigned int v4u __attribute__((ext_vector_type(4)));
typedef int v8i __attribute__((ext_vector_type(8)));
typedef int v4i __attribute__((ext_vector_type(4)));

// Problem constants
#define NB   128   // nodes per graph
#define FIN  7
#define HID  64
#define OUTC 32

// ---- LDS layout (byte offsets). Dynamic shared, total 135744 B ----
#define LDA   132            // padded f32 stride for A rows (128 + 4 via TDM pad)
#define OFF_A    0u          // 128*132*4 = 67584
#define OFF_XT   67584u      // Xt 16x132  = 8448   (dead after phase A)
#define LDXT  132
#define OFF_AX   76032u      // AX 128x20  = 10240  (dead after phase B)
#define LDAX  20
#define OFF_GT   67584u      // Gt 32x132  = 16896  (overlays Xt/AX region)
#define LDGT  132
#define OFF_H1   86272u      // h1 128x68  = 34816
#define LDH1  68
#define OFF_W1T  121088u     // 64x20*4 = 5120
#define LDW1T 20
#define OFF_W2T  126208u     // 32x68*4 = 8704
#define LDW2T 68
#define OFF_B1   134912u
#define OFF_B2   135168u
#define OFF_HS   135296u     // column sums (32 f32)
#define OFF_HP   135424u     // pooled h (32 f32)
#define OFF_T3   135552u     // head tmp (32 f32)
#define OFF_T4   135680u     // head tmp (16 f32)
#define SMEM_BYTES 135744u

// D = A(16xK) * B(Kx16) + C with f32 WMMA, K stepped by 4.
// Ab/Bb are per-lane base pointers already including the lane's row and the
// hi-half K offset (lanes 16..31 read K+2,K+3 per the 16x4 f32 layout).
__device__ __forceinline__ v8f wmma_k(const float* Ab, const float* Bb, int K, v8f c) {
  #pragma unroll 8
  for (int k = 0; k < K; k += 4) {
    v2f a = *(const v2f*)(Ab + k);
    v2f b = *(const v2f*)(Bb + k);
    c = __builtin_amdgcn_wmma_f32_16x16x4_f32(false, a, false, b, (short)0, c,
                                              false, false);
  }
  return c;
}

__global__ void __launch_bounds__(256)
gcn_fused(const float* __restrict__ X,  const float* __restrict__ A,
          const float* __restrict__ W1, const float* __restrict__ b1,
          const float* __restrict__ W2, const float* __restrict__ b2,
          const float* __restrict__ W3, const float* __restrict__ b3,
          const float* __restrict__ W4, const float* __restrict__ b4,
          const float* __restrict__ W5, const float* __restrict__ b5,
          float* __restrict__ out)
{
  extern __shared__ unsigned char smem[];
  float* sA   = (float*)(smem + OFF_A);
  float* sXt  = (float*)(smem + OFF_XT);
  float* sAX  = (float*)(smem + OFF_AX);
  float* sGt  = (float*)(smem + OFF_GT);
  float* sH1  = (float*)(smem + OFF_H1);
  float* sW1t = (float*)(smem + OFF_W1T);
  float* sW2t = (float*)(smem + OFF_W2T);
  float* sB1  = (float*)(smem + OFF_B1);
  float* sB2  = (float*)(smem + OFF_B2);
  float* sHS  = (float*)(smem + OFF_HS);
  float* sHP  = (float*)(smem + OFF_HP);
  float* sT3  = (float*)(smem + OFF_T3);
  float* sT4  = (float*)(smem + OFF_T4);

  const int tid  = threadIdx.x;
  const int wv   = tid >> 5;        // wave id (wave32)
  const int lane = tid & 31;
  const int hi   = lane >> 4;       // 0: K,K+1 half, 1: K+2,K+3 half
  const int lm   = lane & 15;
  const int bidx = blockIdx.x;

  // ---- TDM: DMA A[bidx] (128x128 f32) -> LDS, padding each 128-dword row by
  //      4 dwords so the LDS stride is 132 (bank-conflict friendly).
  if (wv == 0) {
    uint64_t ga  = (uint64_t)(uintptr_t)(A + (size_t)bidx * (NB * NB));
    uint32_t lds = (uint32_t)(uintptr_t)smem + OFF_A;
    v4u g0;
    g0.x = 1u;                                              // count=1 (valid user D#)
    g0.y = lds;                                             // lds_addr
    g0.z = (uint32_t)ga;                                    // global_addr[31:0]
    g0.w = (uint32_t)((ga >> 32) & 0x1FFFFFFu) | (2u << 30); // addr[56:32] | type=2
    v8i g1;
    g1[0] = (int)((2u << 16)      // data_size = 4 bytes
                | (1u << 20)      // pad_enable
                | (6u << 22)      // pad_interval: 2^(6+1)*... = 128 dwords
                | (3u << 25));    // pad_amount: 3 -> 4 dwords
    g1[1] = (int)(128u << 16);    // tensor_dim0[15:0] = 128
    g1[2] = (int)(128u << 16);    // tensor_dim1[15:0] = 128
    g1[3] = (int)(128u << 16);    // tile_dim0 = 128
    g1[4] = 128;                  // tile_dim1 = 128
    g1[5] = 128;                  // tensor_dim0_stride = 128
    g1[6] = 0;
    g1[7] = 0;
    v4i z4 = {0, 0, 0, 0};
#if defined(__clang_major__) && (__clang_major__ >= 23)
    v8i z8 = {0, 0, 0, 0, 0, 0, 0, 0};
    __builtin_amdgcn_tensor_load_to_lds(g0, g1, z4, z4, z8, 0);
#else
    __builtin_amdgcn_tensor_load_to_lds(g0, g1, z4, z4, 0);
#endif
  }

  // ---- Cooperative staging (overlaps with TDM in flight) ----
  {
    const float* Xb = X + (size_t)bidx * (NB * FIN);
    for (int i = tid; i < 16 * NB; i += 256) {        // Xt[n][k] = X[k][n], n>=7 -> 0
      int n = i >> 7, k = i & 127;
      sXt[n * LDXT + k] = (n < FIN) ? Xb[k * FIN + n] : 0.f;
    }
  }
  for (int i = tid; i < HID * 16; i += 256) {         // W1t[n][k] = W1[k][n], k>=7 -> 0
    int n = i >> 4, k = i & 15;
    sW1t[n * LDW1T + k] = (k < FIN) ? W1[k * HID + n] : 0.f;
  }
  for (int i = tid; i < OUTC * HID; i += 256) {       // W2t[n][k] = W2[k][n]
    int n = i >> 6, k = i & 63;
    sW2t[n * LDW2T + k] = W2[k * OUTC + n];
  }
  if (tid < HID)  sB1[tid] = b1[tid];
  if (tid < OUTC) { sB2[tid] = b2[tid]; sHS[tid] = 0.f; }

  if (wv == 0) __builtin_amdgcn_s_wait_tensorcnt(0);
  __syncthreads();

  const int M0  = wv * 16;
  const int row = M0 + 8 * hi;

  // ---- Phase A: AX = A @ Xpad   (M=128, N=16, K=128) ----
  {
    const float* Ab = sA  + (M0 + lm) * LDA  + 2 * hi;
    const float* Bb = sXt + lm * LDXT        + 2 * hi;
    v8f c = {0.f, 0.f, 0.f, 0.f, 0.f, 0.f, 0.f, 0.f};
    c = wmma_k(Ab, Bb, 128, c);
    #pragma unroll
    for (int j = 0; j < 8; ++j)
      sAX[(row + j) * LDAX + lm] = c[j];
  }
  __syncthreads();

  // ---- Phase B: h1 = relu(AX @ W1pad + b1)   (M=128, N=64, K=16) ----
  {
    const float* Ab = sAX + (M0 + lm) * LDAX + 2 * hi;
    #pragma unroll
    for (int nt = 0; nt < 4; ++nt) {
      const int N0 = nt * 16;
      const float* Bb = sW1t + (N0 + lm) * LDW1T + 2 * hi;
      v8f c = {0.f, 0.f, 0.f, 0.f, 0.f, 0.f, 0.f, 0.f};
      c = wmma_k(Ab, Bb, 16, c);
      const float bias = sB1[N0 + lm];
      #pragma unroll
      for (int j = 0; j < 8; ++j)
        sH1[(row + j) * LDH1 + N0 + lm] = fmaxf(c[j] + bias, 0.f);
    }
  }
  __syncthreads();

  // ---- Phase C: G = h1 @ W2 (stored transposed)   (M=128, N=32, K=64) ----
  {
    const float* Ab = sH1 + (M0 + lm) * LDH1 + 2 * hi;
    #pragma unroll
    for (int nt = 0; nt < 2; ++nt) {
      const int N0 = nt * 16;
      const float* Bb = sW2t + (N0 + lm) * LDW2T + 2 * hi;
      v8f c = {0.f, 0.f, 0.f, 0.f, 0.f, 0.f, 0.f, 0.f};
      c = wmma_k(Ab, Bb, 64, c);
      const int col = N0 + lm;
      #pragma unroll
      for (int j = 0; j < 8; ++j)
        sGt[col * LDGT + row + j] = c[j];   // Gt[n][m] = G[m][n]
    }
  }
  __syncthreads();

  // ---- Phase D: h2 = relu(A @ G + b2); mean-pool columns on the fly ----
  {
    const float* Ab = sA + (M0 + lm) * LDA + 2 * hi;
    #pragma unroll
    for (int nt = 0; nt < 2; ++nt) {
      const int N0 = nt * 16;
      const float* Bb = sGt + (N0 + lm) * LDGT + 2 * hi;
      v8f c = {0.f, 0.f, 0.f, 0.f, 0.f, 0.f, 0.f, 0.f};
      c = wmma_k(Ab, Bb, 128, c);
      const int n = N0 + lm;
      const float bias = sB2[n];
      float part = 0.f;
      #pragma unroll
      for (int j = 0; j < 8; ++j)
        part += fmaxf(c[j] + bias, 0.f);
      atomicAdd(&sHS[n], part);             // ds_add_f32
    }
  }
  __syncthreads();

  // ---- Head MLP: (32) -> relu(32) -> relu(16) -> 1 ----
  if (tid < OUTC) sHP[tid] = sHS[tid] * (1.0f / (float)NB);
  __syncthreads();
  if (tid < 32) {
    float acc = b3[tid];
    #pragma unroll
    for (int k = 0; k < 32; ++k) acc += sHP[k] * W3[k * 32 + tid];
    sT3[tid] = fmaxf(acc, 0.f);
  }
  __syncthreads();
  if (tid < 16) {
    float acc = b4[tid];
    #pragma unroll
    for (int k = 0; k < 32; ++k) acc += sT3[k] * W4[k * 16 + tid];
    sT4[tid] = fmaxf(acc, 0.f);
  }
  __syncthreads();
  if (tid == 0) {
    float acc = b5[0];
    #pragma unroll
    for (int k = 0; k < 16; ++k) acc += sT4[k] * W5[k];
    out[bidx] = acc;
  }
}

extern "C" void kernel_launch(void* const* d_in, const int* in_sizes, int n_in,
                              void* d_out, int out_size, void* d_ws, size_t ws_size,
                              hipStream_t stream) {
  const float* X  = (const float*)d_in[0];
  const float* A  = (const float*)d_in[1];
  const float* W1 = (const float*)d_in[2];
  const float* b1 = (const float*)d_in[3];
  const float* W2 = (const float*)d_in[4];
  const float* b2 = (const float*)d_in[5];
  const float* W3 = (const float*)d_in[6];
  const float* b3 = (const float*)d_in[7];
  const float* W4 = (const float*)d_in[8];
  const float* b4 = (const float*)d_in[9];
  const float* W5 = (const float*)d_in[10];
  const float* b5 = (const float*)d_in[11];
  float* out = (float*)d_out;

  const int B = in_sizes[0] / (NB * FIN);   // 4096
  gcn_fused<<<B, 256, SMEM_BYTES, stream>>>(X, A, W1, b1, W2, b2,
                                            W3, b3, W4, b4, W5, b5, out);
}